// EcgFilter_57028575756498
// MI455X (gfx1250) — compile-verified
//
#include <hip/hip_runtime.h>

typedef __attribute__((ext_vector_type(2))) float v2f;
typedef __attribute__((ext_vector_type(8))) float v8f;

#define T_LEN   120000
#define EDGE_I  9
#define TE      (T_LEN + 2 * EDGE_I)     // 120018
#define S_WS    120064                   // padded row stride in ws (floats)
#define IIR_CH  2048
#define IIR_W   1536                     // warmup: 0.9875^1536 < 5e-9
#define FIR_NT  151
#define C_LEN   301                      // autocorr length
#define CB_N    384                      // padded c table (offset 32)
#define TT      512                      // time-tile per workgroup
#define SB      (TT + 320)               // staged LDS row length = 832
#define HALO_L  152

// odd (antisymmetric) extension lookup: ext index t maps into x[0..T)
__device__ __forceinline__ float odd_ext_load(const float* __restrict__ x, int t, int T) {
    if (t < 0)       return 2.0f * x[0]     - x[-t];
    else if (t >= T) return 2.0f * x[T - 1] - x[2 * T - 2 - t];
    return x[t];
}

// ---------------- c = autocorrelation of b_fir (301 taps, padded to 384) ------
__global__ void build_c_kernel(const float* __restrict__ b, float* __restrict__ cbuf) {
    int j = threadIdx.x;                 // 0..383
    float s = 0.0f;
    int q = j - 32;                      // cArr index 0..300
    if (q >= 0 && q <= C_LEN - 1) {
        int lag = q - 150;               // -150..150
        int lo = lag < 0 ? -lag : 0;
        int hi = lag > 0 ? (FIR_NT - 1) - lag : (FIR_NT - 1);
        for (int t = lo; t <= hi; ++t) s = fmaf(b[t], b[t + lag], s);
    }
    cbuf[j] = s;
}

// ---------------- IIR notch filtfilt: forward pass (chunked scan) -------------
__global__ void iir_fwd_kernel(const float* __restrict__ x,
                               const float* __restrict__ bn,
                               const float* __restrict__ an,
                               const float* __restrict__ zin,
                               float* __restrict__ yf, int R, int NC) {
    int gid = blockIdx.x * blockDim.x + threadIdx.x;
    if (gid >= R * NC) return;
    int row = gid / NC, c = gid % NC;
    const float* xr = x + (size_t)row * T_LEN;
    float* yfr = yf + (size_t)row * S_WS;
    float b0 = bn[0], b1 = bn[1], b2 = bn[2], a1 = an[1], a2 = an[2];
    int start = c * IIR_CH;
    int end = start + IIR_CH; if (end > TE) end = TE;
    float z1, z2; int i0;
    if (c == 0) {                        // exact init: zi * ext[0]
        float e0 = 2.0f * xr[0] - xr[EDGE_I];
        z1 = zin[0] * e0; z2 = zin[1] * e0; i0 = 0;
    } else {                             // warm-up from zero state
        z1 = 0.0f; z2 = 0.0f; i0 = start - IIR_W;
    }
    for (int i = i0; i < end; ++i) {
        float xn = odd_ext_load(xr, i - EDGE_I, T_LEN);
        float y  = fmaf(b0, xn, z1);
        z1 = fmaf(b1, xn, z2) - a1 * y;
        z2 = b2 * xn - a2 * y;
        if (i >= start) yfr[i] = y;
    }
}

// ---------------- IIR notch filtfilt: backward pass + crop --------------------
__global__ void iir_bwd_kernel(const float* __restrict__ yf,
                               const float* __restrict__ bn,
                               const float* __restrict__ an,
                               const float* __restrict__ zin,
                               float* __restrict__ y1, int R, int NC) {
    int gid = blockIdx.x * blockDim.x + threadIdx.x;
    if (gid >= R * NC) return;
    int row = gid / NC, c = gid % NC;
    const float* yfr = yf + (size_t)row * S_WS;
    float* y1r = y1 + (size_t)row * S_WS;
    float b0 = bn[0], b1 = bn[1], b2 = bn[2], a1 = an[1], a2 = an[2];
    int start = c * IIR_CH;
    int end = start + IIR_CH; if (end > TE) end = TE;
    float z1, z2; int r0;
    if (c == 0) {                        // exact init: zi * yf[-1]
        float u0 = yfr[TE - 1];
        z1 = zin[0] * u0; z2 = zin[1] * u0; r0 = 0;
    } else {
        z1 = 0.0f; z2 = 0.0f; r0 = start - IIR_W;
    }
    for (int r = r0; r < end; ++r) {
        float xn = yfr[TE - 1 - r];      // reversed read
        float y  = fmaf(b0, xn, z1);
        z1 = fmaf(b1, xn, z2) - a1 * y;
        z2 = b2 * xn - a2 * y;
        if (r >= start) {
            int i = TE - 1 - r;          // ext-domain index of this output
            if (i >= EDGE_I && i < TE - EDGE_I) y1r[i - EDGE_I] = y;
        }
    }
}

#define WMMA_F32(A, B, C) \
    __builtin_amdgcn_wmma_f32_16x16x4_f32(false, (A), false, (B), (short)0, (C), false, false)

// ---- FIR filtfilt == symmetric 301-tap correlation, fp32 WMMA 16x16x4 --------
// D[m][n] = sum_k A[m][k] * B[k][n]
//   A[m][k] = c[k-2-m]                  (banded Toeplitz of autocorr, in LDS)
//   B[k][n] = ext_{row n}[t0+tlb-152+k] (signal tile staged in LDS)
// Each wave owns 4 contiguous 16-sample subtiles: one A fragment feeds 4 WMMAs.
__global__ __launch_bounds__(256) void fir_sym_wmma_kernel(
        const float* __restrict__ y1, const float* __restrict__ cbuf,
        float* __restrict__ out) {
    __shared__ float sC[CB_N];
    __shared__ float sB[16 * SB];

    int tid = threadIdx.x;
    int t0 = blockIdx.x * TT;
    int rowbase = blockIdx.y * 16;

    for (int j = tid; j < CB_N; j += 256) sC[j] = cbuf[j];

    // stage 16 rows x SB cols; float4 fast path, odd-extension at boundaries.
    // (t0 multiple of 512, halo offset -152 => idx multiple of 4 => 16B aligned)
    for (int q = tid; q < 16 * (SB / 4); q += 256) {
        int lr = q / (SB / 4);
        int c4 = (q % (SB / 4)) * 4;
        const float* src = y1 + (size_t)(rowbase + lr) * S_WS;
        int idx = t0 - HALO_L + c4;
        float* dst = &sB[lr * SB + c4];
        if (idx >= 0 && idx + 3 < T_LEN) {
            *(float4*)dst = *(const float4*)(src + idx);
        } else {
            dst[0] = odd_ext_load(src, idx,     T_LEN);
            dst[1] = odd_ext_load(src, idx + 1, T_LEN);
            dst[2] = odd_ext_load(src, idx + 2, T_LEN);
            dst[3] = odd_ext_load(src, idx + 3, T_LEN);
        }
    }
    __syncthreads();

    int wave = tid >> 5, lane = tid & 31;
    int M  = lane & 15;                  // A row / B column (signal row) / D col
    int kh = lane >> 4;                  // K half-select within 16x4 fragment
    const float* cA = &sC[32 + 2 * kh - 2 - M];      // indexed by 4*kk
    int tlb = wave * 64;                 // 4 contiguous subtiles per wave
    const float* pBs = &sB[M * SB + tlb + 2 * kh];   // + j*16 + 4*kk

    v8f acc0 = {0.f,0.f,0.f,0.f,0.f,0.f,0.f,0.f};
    v8f acc1 = acc0, acc2 = acc0, acc3 = acc0;

    #pragma unroll 4
    for (int kk = 0; kk < 80; ++kk) {
        v2f a;  a.x  = cA[4 * kk];       a.y  = cA[4 * kk + 1];
        v2f b0; b0.x = pBs[4 * kk];      b0.y = pBs[4 * kk + 1];
        v2f b1; b1.x = pBs[4 * kk + 16]; b1.y = pBs[4 * kk + 17];
        v2f b2; b2.x = pBs[4 * kk + 32]; b2.y = pBs[4 * kk + 33];
        v2f b3; b3.x = pBs[4 * kk + 48]; b3.y = pBs[4 * kk + 49];
        acc0 = WMMA_F32(a, b0, acc0);
        acc1 = WMMA_F32(a, b1, acc1);
        acc2 = WMMA_F32(a, b2, acc2);
        acc3 = WMMA_F32(a, b3, acc3);
    }

    // D layout: lane gives N=row, VGPR v gives M=v+8*kh => 8 consecutive t/lane
    int trow = rowbase + M;
    float* rowp = out + (size_t)trow * T_LEN;
    v8f accs[4] = {acc0, acc1, acc2, acc3};
    #pragma unroll
    for (int j = 0; j < 4; ++j) {
        int tfirst = t0 + tlb + j * 16 + 8 * kh;
        float* dst = rowp + tfirst;
        v8f a = accs[j];
        if (tfirst + 7 < T_LEN) {
            float4 lo = make_float4(a[0], a[1], a[2], a[3]);
            float4 hi = make_float4(a[4], a[5], a[6], a[7]);
            *(float4*)(dst)     = lo;
            *(float4*)(dst + 4) = hi;
        } else {
            for (int v = 0; v < 8; ++v)
                if (tfirst + v < T_LEN) dst[v] = a[v];
        }
    }
}

extern "C" void kernel_launch(void* const* d_in, const int* in_sizes, int n_in,
                              void* d_out, int out_size, void* d_ws, size_t ws_size,
                              hipStream_t stream) {
    const float* x   = (const float*)d_in[0];
    const float* bn  = (const float*)d_in[1];
    const float* an  = (const float*)d_in[2];
    const float* zin = (const float*)d_in[3];
    const float* bf  = (const float*)d_in[4];
    // d_in[5] (zi_fir) provably cannot affect the cropped output: padlen 453 > 2*150
    float* out = (float*)d_out;

    int R = in_sizes[0] / T_LEN;         // 384 rows

    float* yf = (float*)d_ws;                       // forward IIR output (ext domain)
    float* y1 = yf + (size_t)R * S_WS;              // IIR filtfilt result
    float* cb = y1 + (size_t)R * S_WS;              // 384-float autocorr table

    build_c_kernel<<<1, CB_N, 0, stream>>>(bf, cb);

    int NC = (TE + IIR_CH - 1) / IIR_CH;            // 59 chunks per row
    int nthreads = R * NC;
    int nblk = (nthreads + 255) / 256;
    iir_fwd_kernel<<<nblk, 256, 0, stream>>>(x,  bn, an, zin, yf, R, NC);
    iir_bwd_kernel<<<nblk, 256, 0, stream>>>(yf, bn, an, zin, y1, R, NC);

    dim3 grid((T_LEN + TT - 1) / TT, R / 16);       // (235, 24)
    fir_sym_wmma_kernel<<<grid, 256, 0, stream>>>(y1, cb, out);
}